// Slot_Merger_Cosine_46986942218269
// MI455X (gfx1250) — compile-verified
//
#include <hip/hip_runtime.h>
#include <math.h>

// MI455X reasoning: B=32768 tiny 9x64 f32 problems, 152MB total traffic ->
// ~6.5us at 23.3 TB/s; ~0.7 GFLOP -> pure bandwidth bound (and the whole
// working set fits in the 192MB L2). Keep f32 precision (threshold compare)
// and use V_WMMA_F32_16X16X4_F32 for both sim = X*X^T and out = A'*X,
// folding the reference's scatter/scale/identity logic into the A' matrix.
// One wave32 per batch item. All WMMA operand reads are unconditional
// (zero-padded LDS) so no exec-mask churn; output goes through LDS for
// fully coalesced float4 stores.

typedef __attribute__((ext_vector_type(2))) float v2f;
typedef __attribute__((ext_vector_type(8))) float v8f;

#define SLOTS 9
#define DIM 64
#define EPSV 1e-8f
#define THR 0.9f
#define WAVES_PER_BLOCK 8
#define XROW 68    /* padded LDS row stride (floats): bank = (4*row+col)%64 */
#define SIMROW 17  /* padded LDS row stride for the 16x16 sim tile */

struct __align__(16) WaveShm {
    float x[16 * XROW];       // rows 0..8 = slots, rows 9..15 = zeros;
                              // reused as the output staging tile at the end
    float sim[16 * SIMROW];   // full 16x16 sim tile (rows >= 9 are junk)
    int   wmask[16];          // 9-bit row masks (0 for rows >= 9)
    int   winner[16];         // winner[m] = max s merging into m, else -1
    int   zmask;              // bitmask of zeroed slots
    int   pad[3];
};

__global__ __launch_bounds__(256) void
slot_merger_cosine_kernel(const float* __restrict__ in,
                          float* __restrict__ out_slots,
                          float* __restrict__ out_mask,
                          int nbatch)
{
    __shared__ WaveShm shm[WAVES_PER_BLOCK];
    const int lane = threadIdx.x & 31;
    const int wave = threadIdx.x >> 5;
    const int b    = blockIdx.x * WAVES_PER_BLOCK + wave;
    const bool active = (b < nbatch);          // uniform per wave
    WaveShm& w = shm[wave];

    const int r = lane & 15;       // A row / B,D column index for this lane
    const int h = lane >> 4;       // wave half selects the K sub-pair

    // ---- stage: coalesced float4 loads -> padded LDS; zero pad rows ----
    if (active) {
        const float4* in4 = (const float4*)(in + (size_t)b * (SLOTS * DIM));
#pragma unroll
        for (int t = 0; t < 5; ++t) {
            int idx = lane + 32 * t;           // 144 float4 = 9x64 floats
            if (idx < 144) {
                float4 v = in4[idx];
                *(float4*)&w.x[(idx >> 4) * XROW + ((idx & 15) << 2)] = v;
            }
        }
        const float4 z4 = make_float4(0.f, 0.f, 0.f, 0.f);
#pragma unroll
        for (int t = 0; t < 4; ++t) {          // zero rows 9..15, cols 0..63
            int idx = lane + 32 * t;           // 112 float4
            if (idx < 112) {
                *(float4*)&w.x[(9 + (idx >> 4)) * XROW + ((idx & 15) << 2)] = z4;
            }
        }
        if (lane < 16) { w.wmask[lane] = 0; w.winner[lane] = -1; }
        if (lane == 0) w.zmask = 0;
    }
    __syncthreads();

    // ---- pass 1: sim = X * X^T, 16 chained V_WMMA_F32_16X16X4_F32 ----
    if (active) {
        v2f a[16];
#pragma unroll
        for (int k = 0; k < 16; ++k)           // unconditional ds_load_b64
            a[k] = *(const v2f*)&w.x[r * XROW + 4 * k + 2 * h];
        v8f acc = {};
#pragma unroll
        for (int k = 0; k < 16; ++k)           // B operand == A per lane
            acc = __builtin_amdgcn_wmma_f32_16x16x4_f32(
                false, a[k], false, a[k], (short)0, acc, false, false);
        // D layout: lane holds sim[j + 8h][r]; spill all 16 rows (no predicate)
#pragma unroll
        for (int j = 0; j < 8; ++j)
            w.sim[(j + 8 * h) * SIMROW + r] = acc[j];
    }
    __syncthreads();

    // ---- mask / count / winner bookkeeping on lanes 0..8 ----
    if (active && lane < SLOTS) {
        const int s = lane;
        float ns = sqrtf(w.sim[s * SIMROW + s]);
        int m = 0;
#pragma unroll
        for (int t = 0; t < SLOTS; ++t) {
            float nt  = sqrtf(w.sim[t * SIMROW + t]);
            float dot = w.sim[s * SIMROW + t];
            // cos > THR  <=>  dot > THR*(ns*nt + eps)   (no divide)
            if (dot > THR * (ns * nt + EPSV)) m |= (1 << t);
        }
        w.wmask[s] = m;
        int cnt    = __popc(m);
        int minidx = (m == 0) ? 0 : (__ffs(m) - 1);
        if (cnt > 1) {
            atomicOr(&w.zmask, m & ~(1 << minidx));  // zero merged dups
            atomicMax(&w.winner[minidx], s);         // last writer wins
        }
    }
    __syncthreads();

    if (active) {
        // slot_mask output
        if (lane < SLOTS) {
            int cnt = __popc(w.wmask[lane]);
            float sm = ((w.zmask >> lane) & 1) ? 0.0f
                                               : (cnt > 0 ? 1.0f : 0.0f);
            out_mask[(size_t)b * SLOTS + lane] = sm;
        }

        // ---- build A' row for this lane's M = r ----
        //   winner[M] >= 0 : row = mask[winner]/(count[winner]+eps)
        //   count[M] <= 1  : row = delta_M (keep original slot)
        //   else           : row = 0
        int wdx  = w.winner[r];
        int cntM = __popc(w.wmask[r]);
        int amask; float scale;
        if (wdx >= 0) {
            amask = w.wmask[wdx];
            scale = 1.0f / ((float)__popc(amask) + EPSV);
        } else if (cntM <= 1) {
            amask = 1 << r;
            scale = 1.0f;
        } else {
            amask = 0;
            scale = 0.0f;
        }

        // ---- pass 2: out = A' * X, 4 column tiles x 4 K-steps ----
        v8f accs[4];
#pragma unroll
        for (int c = 0; c < 4; ++c) {
            v8f acc = {};
#pragma unroll
            for (int k = 0; k < 4; ++k) {
                int t0 = 4 * k + 2 * h;
                v2f av, bv;
                av.x = ((amask >> t0) & 1)       ? scale : 0.0f;
                av.y = ((amask >> (t0 + 1)) & 1) ? scale : 0.0f;
                bv.x = w.x[t0 * XROW + c * 16 + r];        // rows >=9 are 0.0
                bv.y = w.x[(t0 + 1) * XROW + c * 16 + r];
                acc = __builtin_amdgcn_wmma_f32_16x16x4_f32(
                    false, av, false, bv, (short)0, acc, false, false);
            }
            accs[c] = acc;
        }

        // spill D tiles into the (now dead) X region: all rows, no predicate.
        // LDS ops are in-order within a wave, so reads above complete first.
#pragma unroll
        for (int c = 0; c < 4; ++c)
#pragma unroll
            for (int j = 0; j < 8; ++j)
                w.x[(j + 8 * h) * XROW + c * 16 + r] = accs[c][j];

        // coalesced float4 readout of rows 0..8
        float4* out4 = (float4*)(out_slots + (size_t)b * (SLOTS * DIM));
#pragma unroll
        for (int t = 0; t < 5; ++t) {
            int idx = lane + 32 * t;
            if (idx < 144) {
                out4[idx] =
                    *(const float4*)&w.x[(idx >> 4) * XROW + ((idx & 15) << 2)];
            }
        }
    }
}

extern "C" void kernel_launch(void* const* d_in, const int* in_sizes, int n_in,
                              void* d_out, int out_size, void* d_ws, size_t ws_size,
                              hipStream_t stream) {
    const float* slots = (const float*)d_in[0];
    const int nbatch = in_sizes[0] / (SLOTS * DIM);
    float* out_slots = (float*)d_out;
    float* out_mask  = (float*)d_out + (size_t)nbatch * SLOTS * DIM;
    const int blocks = (nbatch + WAVES_PER_BLOCK - 1) / WAVES_PER_BLOCK;
    hipLaunchKernelGGL(slot_merger_cosine_kernel, dim3(blocks), dim3(256), 0,
                       stream, slots, out_slots, out_mask, nbatch);
}